// Net_60189671686197
// MI455X (gfx1250) — compile-verified
//
#include <hip/hip_runtime.h>
#include <hip/hip_bf16.h>

typedef __attribute__((ext_vector_type(2))) float v2f;
typedef __attribute__((ext_vector_type(8))) float v8f;

// ---------------------------------------------------------------------------
// Degree / normalization
// ---------------------------------------------------------------------------
__global__ void deg_init_kernel(float* __restrict__ deg, int n) {
    int i = blockIdx.x * blockDim.x + threadIdx.x;
    if (i < n) deg[i] = 1.0f;  // self loop
}

__global__ void deg_count_kernel(const int* __restrict__ dst, float* __restrict__ deg, int e) {
    int i = blockIdx.x * blockDim.x + threadIdx.x;
    if (i < e) atomicAdd(&deg[dst[i]], 1.0f);
}

__global__ void deg_rsqrt_kernel(float* __restrict__ deg, int n) {
    int i = blockIdx.x * blockDim.x + threadIdx.x;
    if (i < n) deg[i] = rsqrtf(deg[i]);
}

// ---------------------------------------------------------------------------
// Fused GEMM + self-loop/bias init:
//   T[i,:] = (relu?)H[i,:] @ W          (message source for scatter)
//   O[i,:] = T[i,:]*dinv[i]^2 + bias    (aggregation accumulator, fully init'd)
// Block: 128 threads = 4 waves. Block covers a 64-row strip; each wave owns one
// 16-col tile and iterates 4 row-tiles (64 x v_wmma_f32_16x16x4_f32 per wave).
// B-fragments hoisted to registers and reused across row-tiles.
// ---------------------------------------------------------------------------
__global__ __launch_bounds__(128) void gemm64_wmma_kernel(
    const float* __restrict__ H, const float* __restrict__ W,
    const float* __restrict__ dinv, const float* __restrict__ bias,
    float* __restrict__ T, float* __restrict__ O, int relu_in, int N) {
    __shared__ float sW[64 * 65];  // padded stride: no bank conflicts
    __shared__ float sA[64 * 65];

    const int tid  = threadIdx.x;
    const int row0 = blockIdx.x * 64;

    // Cooperative load of W (64x64, 1024 float4) into LDS.
    for (int i = tid; i < 1024; i += 128) {
        int r = i >> 4;           // i / 16
        int c = (i & 15) << 2;    // (i % 16) * 4
        float4 v = reinterpret_cast<const float4*>(W)[i];
        sW[r * 65 + c + 0] = v.x;
        sW[r * 65 + c + 1] = v.y;
        sW[r * 65 + c + 2] = v.z;
        sW[r * 65 + c + 3] = v.w;
    }
    // Cooperative load of the 64x64 A strip (1024 float4), fusing previous ReLU.
    for (int i = tid; i < 1024; i += 128) {
        int r = i >> 4;
        int c = (i & 15) << 2;
        int row = row0 + r;
        float4 v = make_float4(0.f, 0.f, 0.f, 0.f);
        if (row < N) v = reinterpret_cast<const float4*>(H + (size_t)row * 64)[c >> 2];
        if (relu_in) {
            v.x = fmaxf(v.x, 0.f); v.y = fmaxf(v.y, 0.f);
            v.z = fmaxf(v.z, 0.f); v.w = fmaxf(v.w, 0.f);
        }
        sA[r * 65 + c + 0] = v.x;
        sA[r * 65 + c + 1] = v.y;
        sA[r * 65 + c + 2] = v.z;
        sA[r * 65 + c + 3] = v.w;
    }
    __syncthreads();

    const int wave = tid >> 5;
    const int lane = tid & 31;
    const int col0 = wave * 16;
    const int half = lane >> 4;   // 0: lanes 0-15, 1: lanes 16-31
    const int l    = lane & 15;

    // Hoist all 16 B-fragments (K=64 in steps of 4) into registers.
    v2f bf[16];
#pragma unroll
    for (int kk = 0; kk < 16; ++kk) {
        const int ka = kk * 4 + 2 * half;  // lane-half selects K pair per ISA layout
        bf[kk].x = sW[ka * 65 + col0 + l];
        bf[kk].y = sW[(ka + 1) * 65 + col0 + l];
    }
    const float bval = bias[col0 + l];

#pragma unroll
    for (int rt = 0; rt < 4; ++rt) {
        const float* aRow = &sA[(rt * 16 + l) * 65];
        v8f acc = {};
#pragma unroll
        for (int kk = 0; kk < 16; ++kk) {
            const int ka = kk * 4 + 2 * half;
            v2f a;
            a.x = aRow[ka];
            a.y = aRow[ka + 1];
            acc = __builtin_amdgcn_wmma_f32_16x16x4_f32(
                /*neg_a=*/false, a, /*neg_b=*/false, bf[kk],
                /*c_mod=*/(short)0, acc, /*reuse_a=*/false, /*reuse_b=*/false);
        }
        // C/D layout: lanes 0-15 -> rows r, lanes 16-31 -> rows r+8.
        const int rbase = row0 + rt * 16 + 8 * half;
#pragma unroll
        for (int r = 0; r < 8; ++r) {
            int rr = rbase + r;
            if (rr < N) {
                float val = acc[r];
                size_t off = (size_t)rr * 64 + col0 + l;
                T[off] = val;
                float di = dinv[rr];
                O[off] = val * di * di + bval;
            }
        }
    }
}

// ---------------------------------------------------------------------------
// Edge scatter: one wave (32 lanes) per edge, 2 features per lane (float2 gather).
// out[dst] += T[src] * dinv[src]*dinv[dst]
// ---------------------------------------------------------------------------
__global__ __launch_bounds__(256) void scatter64_kernel(
    const float* __restrict__ T, const int* __restrict__ src, const int* __restrict__ dst,
    const float* __restrict__ dinv, float* __restrict__ out, int E) {
    int gwave = (blockIdx.x * blockDim.x + threadIdx.x) >> 5;
    int lane  = threadIdx.x & 31;
    if (gwave >= E) return;
    int s = src[gwave];
    int d = dst[gwave];
    float nrm = dinv[s] * dinv[d];
    float2 v = reinterpret_cast<const float2*>(T + (size_t)s * 64)[lane];
    float* o = out + (size_t)d * 64 + 2 * lane;
    atomicAdd(o + 0, v.x * nrm);
    atomicAdd(o + 1, v.y * nrm);
}

// ---------------------------------------------------------------------------
// Final layer: ytmp[i] = dot(relu(h[i]), W_o)   (one wave per node)
// ---------------------------------------------------------------------------
__global__ __launch_bounds__(256) void out_dot_kernel(
    const float* __restrict__ H, const float* __restrict__ Wo,
    float* __restrict__ ytmp, int N) {
    int node = (blockIdx.x * blockDim.x + threadIdx.x) >> 5;
    int lane = threadIdx.x & 31;
    if (node >= N) return;
    float2 v = reinterpret_cast<const float2*>(H + (size_t)node * 64)[lane];
    v.x = fmaxf(v.x, 0.f);
    v.y = fmaxf(v.y, 0.f);
    float2 w = reinterpret_cast<const float2*>(Wo)[lane];
    float p = v.x * w.x + v.y * w.y;
#pragma unroll
    for (int off = 16; off > 0; off >>= 1) p += __shfl_xor(p, off, 32);
    if (lane == 0) ytmp[node] = p;
}

__global__ void final_init_kernel(const float* __restrict__ ytmp, const float* __restrict__ dinv,
                                  const float* __restrict__ bo, float* __restrict__ out, int N) {
    int i = blockIdx.x * blockDim.x + threadIdx.x;
    if (i >= N) return;
    float di = dinv[i];
    out[i] = ytmp[i] * di * di + bo[0];
}

__global__ void final_scatter_kernel(const float* __restrict__ ytmp, const int* __restrict__ src,
                                     const int* __restrict__ dst, const float* __restrict__ dinv,
                                     float* __restrict__ out, int E) {
    int i = blockIdx.x * blockDim.x + threadIdx.x;
    if (i >= E) return;
    int s = src[i];
    int d = dst[i];
    atomicAdd(&out[d], ytmp[s] * dinv[s] * dinv[d]);
}

// ---------------------------------------------------------------------------
// Host-side launch
// ---------------------------------------------------------------------------
static inline size_t align256(size_t x) { return (x + 255) & ~size_t(255); }

extern "C" void kernel_launch(void* const* d_in, const int* in_sizes, int n_in,
                              void* d_out, int out_size, void* d_ws, size_t ws_size,
                              hipStream_t stream) {
    const float* x  = (const float*)d_in[0];
    const int*   ei = (const int*)d_in[1];
    const float* Wi = (const float*)d_in[2];
    const float* bi = (const float*)d_in[3];
    const float* Wh = (const float*)d_in[4];
    const float* bh = (const float*)d_in[5];
    const float* Wo = (const float*)d_in[6];
    const float* bo = (const float*)d_in[7];
    float* out = (float*)d_out;

    const int N   = in_sizes[0] / 64;
    const int E   = in_sizes[1] / 2;
    const int NMP = in_sizes[4] / (64 * 64);

    const int* srcI = ei;
    const int* dstI = ei + E;

    // Workspace carve-up
    char* ws = (char*)d_ws;
    float* dinv = (float*)ws;                 ws += align256((size_t)N * sizeof(float));
    float* bufA = (float*)ws;                 ws += align256((size_t)N * 64 * sizeof(float));
    float* bufB = (float*)ws;                 ws += align256((size_t)N * 64 * sizeof(float));
    float* bufC = (float*)ws;                 ws += align256((size_t)N * 64 * sizeof(float));
    (void)ws_size;

    const int TB = 256;
    const int nBlocksN  = (N + TB - 1) / TB;
    const int nBlocksE  = (E + TB - 1) / TB;
    const int nBlocksEw = (E + 7) / 8;        // 8 waves per 256-thread block
    const int nBlocksNw = (N + 7) / 8;
    const int gemmBlocks = (N + 63) / 64;     // 64-row strips

    // Degrees -> dinv (in place)
    deg_init_kernel<<<nBlocksN, TB, 0, stream>>>(dinv, N);
    deg_count_kernel<<<nBlocksE, TB, 0, stream>>>(dstI, dinv, E);
    deg_rsqrt_kernel<<<nBlocksN, TB, 0, stream>>>(dinv, N);

    // Rotation: T is always bufA; output ping-pongs between bufB and bufC.
    float* T = bufA;
    const float* H = x;
    float* O = bufB;
    float* Oother = bufC;

    // Layer 0: input conv (no input relu); GEMM writes both T and init'd O.
    gemm64_wmma_kernel<<<gemmBlocks, 128, 0, stream>>>(H, Wi, dinv, bi, T, O, /*relu_in=*/0, N);
    scatter64_kernel<<<nBlocksEw, TB, 0, stream>>>(T, srcI, dstI, dinv, O, E);

    // Hidden layers (ReLU fused into GEMM A-load)
    for (int l = 0; l < NMP; ++l) {
        H = O;
        float* tmpO = Oother; Oother = O; O = tmpO;
        gemm64_wmma_kernel<<<gemmBlocks, 128, 0, stream>>>(
            H, Wh + (size_t)l * 64 * 64, dinv, bh + (size_t)l * 64, T, O, /*relu_in=*/1, N);
        scatter64_kernel<<<nBlocksEw, TB, 0, stream>>>(T, srcI, dstI, dinv, O, E);
    }

    // Output conv 64 -> 1 (ReLU fused into dot)
    float* ytmp = bufA;  // T is dead now
    out_dot_kernel<<<nBlocksNw, TB, 0, stream>>>(O, Wo, ytmp, N);
    final_init_kernel<<<nBlocksN, TB, 0, stream>>>(ytmp, dinv, bo, out, N);
    final_scatter_kernel<<<nBlocksE, TB, 0, stream>>>(ytmp, srcI, dstI, dinv, out, E);
}